// Simulation_11407433138282
// MI455X (gfx1250) — compile-verified
//
#include <hip/hip_runtime.h>

#define HIDDEN 20
#define HPAIRS (HIDDEN / 2)
#define BLOCK  256
#define DEPTH  16   // async pipeline depth (power of 2, <= 63 for ASYNCcnt)
#define USE_PK_ASM 1

#if defined(__has_builtin)
#  if __has_builtin(__builtin_amdgcn_global_load_async_to_lds_b32)
#    define HAVE_ASYNC_LDS 1
#  else
#    define HAVE_ASYNC_LDS 0
#  endif
#else
#  define HAVE_ASYNC_LDS 0
#endif

typedef __attribute__((address_space(1))) int as1_int;
typedef __attribute__((address_space(3))) int as3_int;
typedef float vfloat2 __attribute__((ext_vector_type(2)));

// Packed f32 FMA (VOP3P V_PK_FMA_F32): 2 f32 FMAs per issue slot.
// Clang won't select it from generic <2 x float> IR on gfx1250, so emit directly.
// NOTE: all-VGPR operands — the gfx1250 assembler rejects SGPR-pair sources
// for packed-f32 VOP3P ("invalid op_sel operand"); the weight copies into
// VGPRs are hoisted once and amortized over the 1000-step loop.
static __device__ __forceinline__ vfloat2 pk_fma(vfloat2 a, vfloat2 b, vfloat2 c) {
#if USE_PK_ASM
    vfloat2 d;
    asm("v_pk_fma_f32 %0, %1, %2, %3" : "=v"(d) : "v"(a), "v"(b), "v"(c));
    return d;
#else
    return __builtin_elementwise_fma(a, b, c);
#endif
}

__global__ __launch_bounds__(BLOCK) void sim_scan_kernel(
    const float* __restrict__ inputs,   // [T, N]
    const float* __restrict__ x0,       // [N, 2]
    const float* __restrict__ Amat,     // [2,2] row-major
    const float* __restrict__ Bmat,     // [2,1]
    const float* __restrict__ W1,       // [20,1]
    const float* __restrict__ b1,       // [20]
    const float* __restrict__ W2,       // [1,20]
    const float* __restrict__ b2,       // [1]
    float* __restrict__ outp,           // [T, N]
    float* __restrict__ states,         // [T, N, 2]
    int T, int N)
{
    const int tid = threadIdx.x;
    const int n   = blockIdx.x * BLOCK + tid;
#if HAVE_ASYNC_LDS
    __shared__ float sbuf[DEPTH][BLOCK];
#endif
    if (n >= N) return;

    // Wave-uniform parameters -> scalar loads
    const float a00 = Amat[0], a01 = Amat[1], a10 = Amat[2], a11 = Amat[3];
    const float bI0 = Bmat[0], bI1 = Bmat[1];
    const float u_bias = b2[0];
    vfloat2 w1v[HPAIRS], b1v[HPAIRS], w2v[HPAIRS];
#pragma unroll
    for (int j = 0; j < HPAIRS; ++j) {
        vfloat2 a, b, c;
        a.x = W1[2 * j]; a.y = W1[2 * j + 1];
        b.x = b1[2 * j]; b.y = b1[2 * j + 1];
        c.x = W2[2 * j]; c.y = W2[2 * j + 1];
        w1v[j] = a; b1v[j] = b; w2v[j] = c;
    }

    float s0 = x0[2 * n + 0];
    float s1 = x0[2 * n + 1];
    float ratio = s1 / s0;   // ratio of current x; out[t] = ratio of x_{t+1} = next step's ratio

    vfloat2* const statev = (vfloat2*)states;
    unsigned idx = (unsigned)n;   // element index in the [T,N] plane; fits 32-bit (T*N < 2^31)

    // One simulation step: MLP control + state update + NT streaming stores.
    auto step = [&](float inp) {
        const float nin = inp - ratio;
        vfloat2 ninv; ninv.x = nin; ninv.y = nin;
        vfloat2 zero; zero.x = 0.0f; zero.y = 0.0f;
        vfloat2 acc0 = zero, acc1 = zero;   // 2 accumulators: fc2 dep-chain 20 -> 5
#pragma unroll
        for (int j = 0; j < HPAIRS; j += 2) {
            vfloat2 h0 = pk_fma(ninv, w1v[j],     b1v[j]);
            vfloat2 h1 = pk_fma(ninv, w1v[j + 1], b1v[j + 1]);
            // no packed f32 max on CDNA5 (ISA 15.10) -> scalar relu, VOPD-pairable
            h0.x = fmaxf(h0.x, 0.0f); h0.y = fmaxf(h0.y, 0.0f);
            h1.x = fmaxf(h1.x, 0.0f); h1.y = fmaxf(h1.y, 0.0f);
            acc0 = pk_fma(h0, w2v[j],     acc0);
            acc1 = pk_fma(h1, w2v[j + 1], acc1);
        }
        const vfloat2 accs = acc0 + acc1;
        const float u = u_bias + (accs.x + accs.y);

        const float ns0 = __builtin_fmaf(a00, s0, __builtin_fmaf(a01, s1, bI0 * u));
        const float ns1 = __builtin_fmaf(a10, s0, __builtin_fmaf(a11, s1, bI1 * u));
        ratio = ns1 / ns0;
        s0 = ns0; s1 = ns1;

        __builtin_nontemporal_store(ratio, outp + idx);   // 786MB pure write streams:
        vfloat2 st; st.x = ns0; st.y = ns1;               // TH=NT, keep out of L2
        __builtin_nontemporal_store(st, statev + idx);
        idx += (unsigned)N;
    };

#if HAVE_ASYNC_LDS
    if (T >= DEPTH) {
        // ---- CDNA5 async tensor-path input staging: DEPTH-deep ASYNCcnt pipeline ----
#pragma unroll
        for (int p = 0; p < DEPTH; ++p) {
            __builtin_amdgcn_global_load_async_to_lds_b32(
                (as1_int*)(inputs + (unsigned)(n + p * N)), (as3_int*)(&sbuf[p][tid]), 0, 0);
        }
        int t = 0;
        const int mainT = T - DEPTH;
        for (; t < mainT; ++t) {
            // DEPTH loads in flight; async loads complete in order, so waiting for
            // <= DEPTH-1 outstanding guarantees load t has landed in LDS.
            asm volatile("s_wait_asynccnt %0" :: "i"(DEPTH - 1) : "memory");
            const float inp = sbuf[t & (DEPTH - 1)][tid];
            __builtin_amdgcn_global_load_async_to_lds_b32(
                (as1_int*)(inputs + (idx + (unsigned)(DEPTH * N))),
                (as3_int*)(&sbuf[t & (DEPTH - 1)][tid]), 0, 0);
            step(inp);
        }
        for (; t < T; ++t) {  // drain: everything already issued
            asm volatile("s_wait_asynccnt 0" ::: "memory");
            step(sbuf[t & (DEPTH - 1)][tid]);
        }
    } else {
        for (int t = 0; t < T; ++t) step(inputs[idx]);
    }
#else
    // ---- Fallback: direct loads + gfx1250 global_prefetch_b8 look-ahead ----
    for (int t = 0; t < T; ++t) {
        __builtin_prefetch(inputs + (idx + (unsigned)(DEPTH * N)), 0, 0);
        step(inputs[idx]);
    }
#endif
}

extern "C" void kernel_launch(void* const* d_in, const int* in_sizes, int n_in,
                              void* d_out, int out_size, void* d_ws, size_t ws_size,
                              hipStream_t stream) {
    const float* inputs = (const float*)d_in[0];
    const float* x0     = (const float*)d_in[1];
    const float* Amat   = (const float*)d_in[2];
    const float* Bmat   = (const float*)d_in[3];
    const float* W1     = (const float*)d_in[4];
    const float* b1     = (const float*)d_in[5];
    const float* W2     = (const float*)d_in[6];
    const float* b2     = (const float*)d_in[7];

    const int N = in_sizes[1] / 2;          // x0 is [N,2]
    const int T = in_sizes[0] / N;          // inputs is [T,N,1]

    float* outp   = (float*)d_out;              // outputs [T,N,1]
    float* states = outp + (size_t)T * N;       // states  [T,N,2]

    dim3 grid((N + BLOCK - 1) / BLOCK), block(BLOCK);
    sim_scan_kernel<<<grid, block, 0, stream>>>(inputs, x0, Amat, Bmat,
                                                W1, b1, W2, b2,
                                                outp, states, T, N);
}